// RetinaDecoder_39350490366620
// MI455X (gfx1250) — compile-verified
//
#include <hip/hip_runtime.h>
#include <stdint.h>

// RetinaNet-style decoder for MI455X (gfx1250, wave32).
// Roofline: cls_heads stream = 256 MB @ 23.3 TB/s ~= 11 us -> K1 dominates.
// CDNA5 features used:
//  - GLOBAL_LOAD_ASYNC_TO_LDS_B128 + s_wait_asynccnt for the 256 MB stream (K1)
//  - 320 KB LDS: 256 KB 64K-bin radix-select histogram (K2), 128 KB NMS bit-matrix (K3)
// WMMA intentionally unused: this workload contains no matrix multiply.

#define BB 8
#define AA 100000
#define CC 80
#define TOPN 1000
#define MAXO 100
#define CAND 2048
#define MIN_SCORE 0.05f
#define NMS_THR 0.5f

#if defined(__HIP_DEVICE_COMPILE__) && __has_builtin(__builtin_amdgcn_global_load_async_to_lds_b128)
#define ASYNC_OK 1
#else
#define ASYNC_OK 0
#endif

#if ASYNC_OK
typedef int v4i_cdna5 __attribute__((ext_vector_type(4)));
typedef __attribute__((address_space(1))) v4i_cdna5* gptr_v4i;  // global
typedef __attribute__((address_space(3))) v4i_cdna5* lptr_v4i;  // LDS
#endif

__device__ __forceinline__ void wait_async0() {
#if defined(__HIP_DEVICE_COMPILE__)
#if __has_builtin(__builtin_amdgcn_s_wait_asynccnt)
  __builtin_amdgcn_s_wait_asynccnt(0);
#else
  asm volatile("s_wait_asynccnt 0" ::: "memory");
#endif
#endif
}

// ---------------------------------------------------------------------------
// K1: per-anchor class max/argmax (first-max, like jnp.argmax) + box decode.
// One block = 256 anchors. cls tile (80 KB) staged via async copies to LDS so
// global loads are perfectly coalesced (lane-consecutive 16B chunks).
// ---------------------------------------------------------------------------
__global__ __launch_bounds__(256) void k1_reduce_decode(
    const float* __restrict__ cls, const float* __restrict__ reg,
    const float* __restrict__ anc, float* __restrict__ scores,
    int* __restrict__ classes, float* __restrict__ boxes) {
  const int tid = threadIdx.x;
  const int g0  = blockIdx.x * 256;
  const int g   = g0 + tid;  // flat anchor id over B*A (grid sized exactly)

  float best = -3.0e38f;
  int   bi   = 0;

#if ASYNC_OK
  extern __shared__ unsigned char smem1[];
  float* tile = (float*)smem1;  // 256*80 floats = 80 KB
  float* gsrc = const_cast<float*>(cls) + (size_t)g0 * CC;
#pragma unroll
  for (int it = 0; it < 20; ++it) {
    const int off = (it * 256 + tid) * 4;  // float index; 16B per lane per issue
    __builtin_amdgcn_global_load_async_to_lds_b128(
        (gptr_v4i)(gsrc + off), (lptr_v4i)(tile + off), 0, 0);
  }
  wait_async0();     // own wave's async copies done
  __syncthreads();   // all waves' copies done
  const float4* r4 = (const float4*)(tile + tid * CC);
#else
  const float4* r4 = (const float4*)(cls + (size_t)g * CC);
#endif

#pragma unroll
  for (int c = 0; c < CC / 4; ++c) {
    float4 v = r4[c];
    const int base = c * 4;
    if (v.x > best) { best = v.x; bi = base + 0; }  // strict '>' keeps first max
    if (v.y > best) { best = v.y; bi = base + 1; }
    if (v.z > best) { best = v.z; bi = base + 2; }
    if (v.w > best) { best = v.w; bi = base + 3; }
  }
  scores[g]  = best;
  classes[g] = bi;

  const float4 rg = ((const float4*)reg)[g];
  const float4 an = ((const float4*)anc)[g];
  const float aw = an.z - an.x, ah = an.w - an.y;
  const float acx = an.x + 0.5f * aw, acy = an.y + 0.5f * ah;
  const float pw = expf(rg.z) * aw, ph = expf(rg.w) * ah;
  const float pcx = rg.x * aw + acx, pcy = rg.y * ah + acy;
  float4 bb;
  bb.x = truncf(pcx - 0.5f * pw);
  bb.y = truncf(pcy - 0.5f * ph);
  bb.z = truncf(pcx + 0.5f * pw);
  bb.w = truncf(pcy + 0.5f * ph);
  ((float4*)boxes)[g] = bb;
}

// ---------------------------------------------------------------------------
// K2: exact stable top-1000 per image via 2-pass 16-bit radix select using a
// 64K-bin LDS histogram (256 KB — needs CDNA5's 320 KB LDS), then a 2048-wide
// bitonic sort of boundary candidates on key = (score_bits<<32) | ~index,
// which reproduces jax.lax.top_k order exactly (score desc, index asc ties).
// ---------------------------------------------------------------------------
__global__ __launch_bounds__(1024) void k2_topk(
    const float* __restrict__ scores, float* __restrict__ topS,
    unsigned* __restrict__ topI) {
  extern __shared__ unsigned char smem2[];
  unsigned* hist = (unsigned*)smem2;                                       // [65536]
  unsigned long long* cand = (unsigned long long*)(smem2 + 65536u * 4u);   // [CAND]
  unsigned* ctl = (unsigned*)(smem2 + 65536u * 4u + CAND * 8u);            // [16]
  unsigned* grp = ctl + 16;                                                // [1024]

  const int img = blockIdx.x;
  const int tid = threadIdx.x;
  const float* sc = scores + (size_t)img * AA;

  // ---- pass 1: histogram of high 16 bits (positive floats sort as uints) ----
  for (int i = tid; i < 65536; i += 1024) hist[i] = 0u;
  __syncthreads();
  for (int a = tid; a < AA; a += 1024) {
    float s = sc[a];
    if (s > MIN_SCORE) atomicAdd(&hist[__float_as_uint(s) >> 16], 1u);
  }
  __syncthreads();
  { unsigned gs = 0; const int b0 = tid * 64;
    for (int k = 0; k < 64; ++k) gs += hist[b0 + k];
    grp[tid] = gs; }
  __syncthreads();
  if (tid == 0) {
    unsigned cum = 0; int gstar = -1;
    for (int gI = 1023; gI >= 0; --gI) {
      if (cum + grp[gI] >= TOPN) { gstar = gI; break; }
      cum += grp[gI];
    }
    if (gstar < 0) {            // fewer than TOPN valid: take everything valid
      ctl[2] = 1u; ctl[4] = 1u;
    } else {
      unsigned H = 0;
      for (int b = 63; b >= 0; --b) {
        unsigned c = hist[gstar * 64 + b];
        if (cum + c >= TOPN) { H = (unsigned)(gstar * 64 + b); break; }
        cum += c;
      }
      ctl[0] = H; ctl[1] = cum; ctl[4] = 0u;  // cum = count(hi > H)
    }
  }
  __syncthreads();

  // ---- pass 2: low 16 bits restricted to hi == H ----
  if (ctl[4] == 0u) {
    const unsigned H = ctl[0];
    for (int i = tid; i < 65536; i += 1024) hist[i] = 0u;
    __syncthreads();
    for (int a = tid; a < AA; a += 1024) {
      float s = sc[a];
      if (s > MIN_SCORE) {
        unsigned b = __float_as_uint(s);
        if ((b >> 16) == H) atomicAdd(&hist[b & 0xFFFFu], 1u);
      }
    }
    __syncthreads();
    { unsigned gs = 0; const int b0 = tid * 64;
      for (int k = 0; k < 64; ++k) gs += hist[b0 + k];
      grp[tid] = gs; }
    __syncthreads();
    if (tid == 0) {
      const unsigned need = TOPN - ctl[1];
      unsigned cum = 0; int gstar = 0;
      for (int gI = 1023; gI >= 0; --gI) {
        if (cum + grp[gI] >= need) { gstar = gI; break; }
        cum += grp[gI];
      }
      unsigned L = 0;
      for (int b = 63; b >= 0; --b) {
        unsigned c = hist[gstar * 64 + b];
        if (cum + c >= need) { L = (unsigned)(gstar * 64 + b); break; }
        cum += c;
      }
      ctl[2] = (H << 16) | L;
    }
    __syncthreads();
  }

  // ---- gather bits >= Kbits (incl. all boundary ties), sort, emit top 1000 ----
  const unsigned Kbits = ctl[2];
  if (tid == 0) ctl[3] = 0u;
  __syncthreads();
  for (int a = tid; a < AA; a += 1024) {
    float s = sc[a];
    if (s > MIN_SCORE) {
      unsigned b = __float_as_uint(s);
      if (b >= Kbits) {
        unsigned pos = atomicAdd(&ctl[3], 1u);
        if (pos < CAND)
          cand[pos] = ((unsigned long long)b << 32) |
                      (unsigned long long)(0xFFFFFFFFu - (unsigned)a);
      }
    }
  }
  __syncthreads();
  unsigned cnt = ctl[3]; if (cnt > CAND) cnt = CAND;
  for (int i = tid; i < CAND; i += 1024)
    if ((unsigned)i >= cnt) cand[i] = 0ull;
  __syncthreads();

  for (unsigned k = 2; k <= CAND; k <<= 1) {          // bitonic, descending
    for (unsigned j = k >> 1; j > 0; j >>= 1) {
      for (unsigned i = (unsigned)tid; i < CAND; i += 1024) {
        unsigned ixj = i ^ j;
        if (ixj > i) {
          unsigned long long a0 = cand[i], b0 = cand[ixj];
          bool sw = ((i & k) == 0) ? (a0 < b0) : (a0 > b0);
          if (sw) { cand[i] = b0; cand[ixj] = a0; }
        }
      }
      __syncthreads();
    }
  }
  for (int i = tid; i < TOPN; i += 1024) {
    unsigned long long kv = cand[i];
    unsigned bits = (unsigned)(kv >> 32);
    topS[img * TOPN + i] = bits ? __uint_as_float(bits) : -3.0e38f;
    topI[img * TOPN + i] = bits ? (0xFFFFFFFFu - (unsigned)(kv & 0xFFFFFFFFull)) : 0u;
  }
}

// ---------------------------------------------------------------------------
// K3: NMS + compaction. Full 1000x1024 suppression bit-matrix in LDS (128 KB),
// one wave32 runs the greedy sweep (lane = column word, shfl broadcast of the
// "row alive" bit), then popcount-prefix compaction of the first 100 kept.
// ---------------------------------------------------------------------------
__global__ __launch_bounds__(1024) void k3_nms(
    const float* __restrict__ boxes, const int* __restrict__ classes,
    const float* __restrict__ topS, const unsigned* __restrict__ topI,
    float* __restrict__ out) {
  extern __shared__ unsigned char smem3[];
  float4*   bx   = (float4*)smem3;                    // [1000]  16000 B
  float*    area = (float*)(smem3 + 16000);           // [1000]
  float*    sL   = area + 1000;                       // [1000]
  float*    cL   = sL + 1000;                         // [1000]
  unsigned* mask = (unsigned*)(cL + 1000);            // [32000] 128000 B
  unsigned* remW = mask + 32000;                      // [32]
  unsigned* pre  = remW + 32;                         // [32]
  float*    oS   = (float*)(pre + 32);                // [100]
  float*    oC   = oS + 100;                          // [100]
  float*    oB   = oC + 100;                          // [400]

  const int img = blockIdx.x;
  const int tid = threadIdx.x;

  for (int i = tid; i < TOPN; i += 1024) {
    float s = topS[img * TOPN + i];
    unsigned idx = topI[img * TOPN + i];
    sL[i] = s;
    cL[i] = (float)classes[(size_t)img * AA + idx];
    float4 b = ((const float4*)boxes)[(size_t)img * AA + idx];
    bx[i] = b;
    area[i] = fmaxf((b.z - b.x) * (b.w - b.y), 1e-4f);
  }
  if (tid < MAXO) {
    oS[tid] = -1.0f; oC[tid] = -1.0f;
    oB[4 * tid + 0] = 0.f; oB[4 * tid + 1] = 0.f;
    oB[4 * tid + 2] = 0.f; oB[4 * tid + 3] = 0.f;
  }
  __syncthreads();

  // suppression bit-matrix: task t = (row i, column word w)
  for (int t = tid; t < TOPN * 32; t += 1024) {
    const int i = t >> 5, w = t & 31;
    const float4 bi = bx[i];
    const float  ai = area[i];
    unsigned m = 0u;
    const int j0 = w << 5;
    for (int u = 0; u < 32; ++u) {
      const int j = j0 + u;
      if (j < TOPN && j > i) {
        float4 bj = bx[j];
        float iw = fminf(bi.z, bj.z) - fmaxf(bi.x, bj.x);
        float ih = fminf(bi.w, bj.w) - fmaxf(bi.y, bj.y);
        float inter = fmaxf(iw, 0.f) * fmaxf(ih, 0.f);
        float uni = fmaxf(ai + area[j] - inter, 1e-4f);
        if (inter / uni >= NMS_THR) m |= (1u << u);
      }
    }
    mask[t] = m;
  }
  __syncthreads();

  if (tid < 32) {  // single wave32 greedy sweep; lane owns column word 'tid'
    unsigned rem = 0u;
    for (int u = 0; u < 32; ++u) {
      const int i = (tid << 5) + u;
      const bool valid = (i < TOPN) && (sL[i] > MIN_SCORE);
      if (!valid) rem |= (1u << u);   // invalid entries start suppressed
    }
    for (int i = 0; i < TOPN; ++i) {
      unsigned word = __shfl(rem, i >> 5, 32);
      if (((word >> (i & 31)) & 1u) == 0u)   // row i still alive -> suppresses
        rem |= mask[(i << 5) + tid];
    }
    remW[tid] = rem;
  }
  __syncthreads();
  if (tid == 0) {
    unsigned run = 0;
    for (int w = 0; w < 32; ++w) { pre[w] = run; run += __popc(~remW[w]); }
  }
  __syncthreads();

  for (int i = tid; i < TOPN; i += 1024) {
    const unsigned w = (unsigned)i >> 5, b = (unsigned)i & 31u;
    const unsigned keptw = ~remW[w];
    if ((keptw >> b) & 1u) {
      const unsigned rank = pre[w] + __popc(keptw & ((1u << b) - 1u));
      if (rank < MAXO) {
        oS[rank] = sL[i];
        oC[rank] = cL[i];
        float4 bbv = bx[i];
        oB[4 * rank + 0] = bbv.x; oB[4 * rank + 1] = bbv.y;
        oB[4 * rank + 2] = bbv.z; oB[4 * rank + 3] = bbv.w;
      }
    }
  }
  __syncthreads();

  // output layout: s[8][100] @0, c[8][100] @800, b[8][100][4] @1600
  if (tid < MAXO) {
    out[img * MAXO + tid]            = oS[tid];
    out[BB * MAXO + img * MAXO + tid] = oC[tid];
    const int bo = 2 * BB * MAXO + (img * MAXO + tid) * 4;
    out[bo + 0] = oB[4 * tid + 0]; out[bo + 1] = oB[4 * tid + 1];
    out[bo + 2] = oB[4 * tid + 2]; out[bo + 3] = oB[4 * tid + 3];
  }
}

// ---------------------------------------------------------------------------
extern "C" void kernel_launch(void* const* d_in, const int* in_sizes, int n_in,
                              void* d_out, int out_size, void* d_ws, size_t ws_size,
                              hipStream_t stream) {
  (void)in_sizes; (void)n_in; (void)out_size; (void)ws_size;
  const float* cls = (const float*)d_in[0];
  const float* reg = (const float*)d_in[1];
  const float* anc = (const float*)d_in[2];
  float* out = (float*)d_out;

  // workspace carve-up (~19.3 MB)
  float*    scores  = (float*)d_ws;                               // [B*A]
  int*      classes = (int*)(scores + (size_t)BB * AA);           // [B*A]
  float*    boxes   = (float*)(classes + (size_t)BB * AA);        // [B*A*4]
  float*    topS    = boxes + (size_t)BB * AA * 4;                // [B*TOPN]
  unsigned* topI    = (unsigned*)(topS + (size_t)BB * TOPN);      // [B*TOPN]

  const int k1_smem = 256 * CC * 4;                               // 80 KB tile
  const int k2_smem = 65536 * 4 + CAND * 8 + (16 + 1024) * 4;     // ~276 KB
  const int k3_smem = 158656;                                     // ~155 KB

  k1_reduce_decode<<<(BB * AA) / 256, 256, k1_smem, stream>>>(
      cls, reg, anc, scores, classes, boxes);
  k2_topk<<<BB, 1024, k2_smem, stream>>>(scores, topS, topI);
  k3_nms<<<BB, 1024, k3_smem, stream>>>(boxes, classes, topS, topI, out);
}